// MultiheadLSHSelfAttention_2052994367987
// MI455X (gfx1250) — compile-verified
//
#include <hip/hip_runtime.h>
#include <math.h>

#define N_  4
#define T_  4096
#define D_  1024
#define H_  16
#define DH_ 64
#define NH_ 2
#define BS_ 64
#define NB_ 64
#define C_  (NH_*NB_)   /* 128 chunks */
#define B_  (N_*H_)     /* 64 folded batch */
#define M_  (N_*T_)     /* 16384 rows */

typedef __attribute__((ext_vector_type(2))) float v2f;
typedef __attribute__((ext_vector_type(8))) float v8f;

// LDS layout for lsh_attn (dword offsets; dynamic smem starts at LDS byte 0).
// Strides padded so WMMA operand fetches are bank-conflict-free and rows stay
// 16B aligned for async b128 writes (68*4=272, 132*4=528: multiples of 16).
#define QS_   68
#define KS_   68
#define VS_   64
#define DS_   132
#define QOFF_ 0
#define KOFF_ (QOFF_ + 64*QS_)     /* 4352  */
#define VOFF_ (KOFF_ + 128*KS_)    /* 13056 */
#define DOFF_ (VOFF_ + 128*VS_)    /* 21248 */
#define LOFF_ (DOFF_ + 64*DS_)     /* 29696 */
#define ATTN_DWORDS (LOFF_ + 64 + 384)

// Async 16B global->LDS copy (GLOBAL_LOAD_ASYNC_TO_LDS_B128, ASYNCcnt).
__device__ __forceinline__ void g2l_b128(unsigned ldsByte, const float* gp) {
  asm volatile("global_load_async_to_lds_b128 %0, %1, off"
               :: "v"(ldsByte), "v"(gp)
               : "memory");
}
__device__ __forceinline__ void wait_async0() {
  asm volatile("s_wait_asynccnt 0" ::: "memory");
}

// ---------------------------------------------------------------------------
// Kernel 1: fused Q = x@Wq + bq, V = x@Wv + bv via V_WMMA_F32_16X16X4_F32.
// Each wave: 16 rows x 64 cols of BOTH Q and V (8 f32 accumulators).
// Output written directly in per-head layout qk[(n*H+h), t, f].
// ---------------------------------------------------------------------------
__global__ __launch_bounds__(256) void qv_gemm_f32_wmma(
    const float* __restrict__ x,  const float* __restrict__ Wq,
    const float* __restrict__ bq, const float* __restrict__ Wv,
    const float* __restrict__ bv, float* __restrict__ outQ,
    float* __restrict__ outV) {
  int wave = threadIdx.x >> 5;
  int lane = threadIdx.x & 31;
  int task = blockIdx.x * 8 + wave;        // 16384 tasks total
  int rowTile = task >> 4;                 // 0..1023  (16-row tiles)
  int strip   = task & 15;                 // 0..15    (64-col strips)
  int m0 = rowTile << 4;
  int c0 = strip  << 6;
  int half = lane >> 4;
  int l15  = lane & 15;

  v8f accQ[4], accV[4];
  v8f vzero = {0.f,0.f,0.f,0.f,0.f,0.f,0.f,0.f};
  #pragma unroll
  for (int t = 0; t < 4; ++t) { accQ[t] = vzero; accV[t] = vzero; }

  // A-matrix (16x4 f32): lane m = lane&15; half selects K {0,1} vs {2,3}.
  const float* xrow = x + (size_t)(m0 + l15) * D_;
  #pragma unroll 4
  for (int kb = 0; kb < D_; kb += 4) {
    v2f a;
    a[0] = xrow[kb + 2*half];
    a[1] = xrow[kb + 2*half + 1];
    #pragma unroll
    for (int tt = 0; tt < 4; ++tt) {
      int n = c0 + (tt << 4) + l15;
      size_t w0 = (size_t)(kb + 2*half) * D_ + n;  // B (4x16): row k, col n
      v2f bqv, bvv;
      bqv[0] = Wq[w0]; bqv[1] = Wq[w0 + D_];
      bvv[0] = Wv[w0]; bvv[1] = Wv[w0 + D_];
      accQ[tt] = __builtin_amdgcn_wmma_f32_16x16x4_f32(
          false, a, false, bqv, (short)0, accQ[tt], false, false);
      accV[tt] = __builtin_amdgcn_wmma_f32_16x16x4_f32(
          false, a, false, bvv, (short)0, accV[tt], false, false);
    }
  }

  // C/D layout: VGPR r -> row m0 + r + 8*half, col = n tile + (lane&15).
  #pragma unroll
  for (int tt = 0; tt < 4; ++tt) {
    int d = c0 + (tt << 4) + l15;
    int h = d >> 6, f = d & 63;
    float biasq = bq[d], biasv = bv[d];
    #pragma unroll
    for (int r = 0; r < 8; ++r) {
      int m = m0 + r + 8*half;
      int t = m & (T_ - 1);
      int nb = m >> 12;                       // m / T_
      size_t o = (((size_t)(nb * H_ + h)) * T_ + t) * DH_ + f;
      outQ[o] = accQ[tt][r] + biasq;
      outV[o] = accV[tt][r] + biasv;
    }
  }
}

// ---------------------------------------------------------------------------
// Kernel 2: LSH hashing.  rotated[b,nh,t,i] = sum_f qk[b,t,f]*rot[h,f,nh,i];
// bucket = argmax over [r, -r] (first-max), + nh*NB.
// ---------------------------------------------------------------------------
__global__ __launch_bounds__(64) void lsh_hash(const float* __restrict__ qk,
                                               const float* __restrict__ rot,
                                               int* __restrict__ buckets) {
  __shared__ float qs[64][65];   // padded: avoid 64-way bank conflict
  __shared__ float rs[64][64];   // rot for this head: [f][nh*32+i]
  int b  = blockIdx.y;
  int t0 = blockIdx.x << 6;
  int tid = threadIdx.x;
  int h = b & (H_ - 1);
  for (int k = 0; k < 64; ++k)
    qs[k][tid] = qk[((size_t)b * T_ + t0 + k) * DH_ + tid];
  for (int k = 0; k < 64; ++k)
    rs[k][tid] = rot[(size_t)(h * DH_ + k) * 64 + tid];  // (nh,i) flat = nh*32+i
  __syncthreads();

  float rv[32];
  for (int nh = 0; nh < NH_; ++nh) {
    #pragma unroll 4
    for (int i = 0; i < 32; ++i) {
      float a = 0.f;
      for (int f = 0; f < 64; ++f) a += qs[tid][f] * rs[f][nh*32 + i];
      rv[i] = a;
    }
    float best = rv[0]; int arg = 0;
    for (int j = 1; j < 64; ++j) {          // scan order == concat([r,-r])
      float vv = (j < 32) ? rv[j] : -rv[j - 32];
      if (vv > best) { best = vv; arg = j; }
    }
    buckets[(size_t)b * (NH_*T_) + nh * T_ + t0 + tid] = arg + nh * NB_;
  }
}

// ---------------------------------------------------------------------------
// Kernel 3: stable counting sort by (bucket, u).  bucket encodes the hash
// round (nh*NB offset); within a bucket ascending u == ascending t, so a
// stable counting sort over u reproduces argsort(T*bucket + t) exactly.
// ---------------------------------------------------------------------------
__global__ __launch_bounds__(64) void bucket_sort(const int* __restrict__ buckets,
                                                  int* __restrict__ sticker,
                                                  int* __restrict__ undo) {
  __shared__ unsigned hist[64][128];
  int b = blockIdx.x, tid = threadIdx.x;
  for (int bin = 0; bin < 128; ++bin) hist[tid][bin] = 0;
  __syncthreads();
  const int* bb = buckets + (size_t)b * (NH_*T_);
  int u0 = tid * 128;
  for (int e = 0; e < 128; ++e) hist[tid][bb[u0 + e]]++;
  __syncthreads();
  if (tid == 0) {                         // serial (bin, chunk) exclusive scan
    unsigned run = 0;
    for (int bin = 0; bin < 128; ++bin)
      for (int t2 = 0; t2 < 64; ++t2) {
        unsigned cnt = hist[t2][bin]; hist[t2][bin] = run; run += cnt;
      }
  }
  __syncthreads();
  int* st = sticker + (size_t)b * (NH_*T_);
  int* ud = undo    + (size_t)b * (NH_*T_);
  for (int e = 0; e < 128; ++e) {         // stable scatter (per-thread bins)
    int u = u0 + e;
    int bin = bb[u];
    unsigned j = hist[tid][bin]++;
    st[j] = u; ud[u] = (int)j;
  }
}

// ---------------------------------------------------------------------------
// Kernel 4: chunked attention.  One workgroup (128 threads = 4 waves) per
// (b, chunk).  Async b128 gathers into ~118KB LDS; q.k^T and p.v both run on
// V_WMMA_F32_16X16X4_F32 (full f32 precision).  Each wave owns one 16-row
// M-tile: dots = 16 K-steps x 8 N-tiles, p.v = 32 K-steps x 4 N-tiles.
// ---------------------------------------------------------------------------
__global__ __launch_bounds__(128) void lsh_attn(
    const float* __restrict__ qk, const float* __restrict__ vv,
    const int* __restrict__ sticker, const int* __restrict__ buckets,
    float* __restrict__ so, float* __restrict__ slog) {
  extern __shared__ float smem[];
  float* q_s   = smem + QOFF_;         // [64][QS_]
  float* k_s   = smem + KOFF_;         // [128][KS_]
  float* v_s   = smem + VOFF_;         // [128][VS_]
  float* dots  = smem + DOFF_;         // [64][DS_]
  float* lse_s = smem + LOFF_;         // 64
  int* qt  = (int*)(lse_s + 64);       // 64
  int* qb2 = qt + 64;                  // 64
  int* kvt = qb2 + 64;                 // 128
  int* kvb = kvt + 128;                // 128

  int c = blockIdx.x, b = blockIdx.y;
  int tid = threadIdx.x;
  int cprev = (c + C_ - 1) & (C_ - 1);
  const int* stb = sticker + (size_t)b * (NH_*T_);
  const int* bkb = buckets + (size_t)b * (NH_*T_);

  if (tid < 64) {
    int u = stb[c * BS_ + tid];
    qt[tid]  = u & (T_ - 1);
    qb2[tid] = bkb[u];
  }
  {
    int jk = (tid < BS_) ? (c * BS_ + tid) : (cprev * BS_ + (tid - BS_)); // look-back
    int u  = stb[jk];
    kvt[tid] = u & (T_ - 1);
    kvb[tid] = bkb[u];
  }
  __syncthreads();

  // --- async scatter-gather: 16B units straight into LDS (ASYNCcnt) ---
  for (int u = tid; u < 64*16; u += 128) {
    int r = u >> 4, seg = (u & 15) << 2;
    g2l_b128((unsigned)(QOFF_ + r*QS_ + seg) * 4u,
             qk + ((size_t)b * T_ + qt[r]) * DH_ + seg);
  }
  for (int u = tid; u < 128*16; u += 128) {
    int r = u >> 4, seg = (u & 15) << 2;
    size_t g = ((size_t)b * T_ + kvt[r]) * DH_ + seg;
    g2l_b128((unsigned)(KOFF_ + r*KS_ + seg) * 4u, qk + g);
    g2l_b128((unsigned)(VOFF_ + r*VS_ + seg) * 4u, vv + g);
  }
  wait_async0();
  __syncthreads();

  {                                       // normalize k row `tid`
    float s = 0.f;
    for (int f = 0; f < 64; ++f) { float kk = k_s[tid*KS_+f]; s += kk*kk; }
    float sc = 1.f / (sqrtf(s) + 1e-6f);
    for (int f = 0; f < 64; ++f) k_s[tid*KS_+f] *= sc;
  }
  __syncthreads();

  int lane = tid & 31, half = lane >> 4, l15 = lane & 15;
  int m0 = (tid >> 5) << 4;               // wave's 16-row M-tile
  v8f vzero = {0.f,0.f,0.f,0.f,0.f,0.f,0.f,0.f};

  // --- dots = q . k^T on WMMA: K=64, N=128 ---
  v8f accD[8];
  #pragma unroll
  for (int t = 0; t < 8; ++t) accD[t] = vzero;
  for (int kb = 0; kb < 64; kb += 4) {
    int kk = kb + 2*half;
    v2f a;
    a[0] = q_s[(m0 + l15)*QS_ + kk];
    a[1] = q_s[(m0 + l15)*QS_ + kk + 1];
    #pragma unroll
    for (int tt = 0; tt < 8; ++tt) {
      int n = (tt << 4) + l15;
      v2f bb;                             // B(k,n) = khat[n][k]
      bb[0] = k_s[n*KS_ + kk];
      bb[1] = k_s[n*KS_ + kk + 1];
      accD[tt] = __builtin_amdgcn_wmma_f32_16x16x4_f32(
          false, a, false, bb, (short)0, accD[tt], false, false);
    }
  }
  // masked store (reference override order: bucket, causal, self)
  #pragma unroll
  for (int tt = 0; tt < 8; ++tt) {
    int n = (tt << 4) + l15;
    int kt = kvt[n], kb2 = kvb[n];
    #pragma unroll
    for (int r = 0; r < 8; ++r) {
      int m = m0 + r + 8*half;
      float d = accD[tt][r] * 0.125f;     // DH^-0.5
      if (qb2[m] != kb2) d = -INFINITY;
      if (kt > qt[m])    d = -INFINITY;
      if (kt == qt[m])   d = -1e-5f;
      dots[m*DS_ + n] = d;
    }
  }
  __syncthreads();

  if (tid < 64) {                         // row LSE + probs
    float m = -INFINITY;
    for (int col = 0; col < 128; ++col) m = fmaxf(m, dots[tid*DS_+col]);
    float s = 0.f;
    for (int col = 0; col < 128; ++col) s += expf(dots[tid*DS_+col] - m);
    float l = m + logf(s);
    lse_s[tid] = l;
    for (int col = 0; col < 128; ++col)
      dots[tid*DS_+col] = expf(dots[tid*DS_+col] - l);
  }
  __syncthreads();

  // --- out = p . v on WMMA: K=128, N=64 ---
  v8f accO[4];
  #pragma unroll
  for (int t = 0; t < 4; ++t) accO[t] = vzero;
  for (int kb = 0; kb < 128; kb += 4) {
    int kk = kb + 2*half;
    v2f a;
    a[0] = dots[(m0 + l15)*DS_ + kk];
    a[1] = dots[(m0 + l15)*DS_ + kk + 1];
    #pragma unroll
    for (int tt = 0; tt < 4; ++tt) {
      int n = (tt << 4) + l15;
      v2f bb;                             // B(k,n) = v_s[k][n]
      bb[0] = v_s[kk*VS_ + n];
      bb[1] = v_s[(kk+1)*VS_ + n];
      accO[tt] = __builtin_amdgcn_wmma_f32_16x16x4_f32(
          false, a, false, bb, (short)0, accO[tt], false, false);
    }
  }
  #pragma unroll
  for (int tt = 0; tt < 4; ++tt) {
    int n = (tt << 4) + l15;
    #pragma unroll
    for (int r = 0; r < 8; ++r) {
      int m = m0 + r + 8*half;
      so[(((size_t)b * (NH_*T_)) + c * BS_ + m) * DH_ + n] = accO[tt][r];
    }
  }
  if (tid < 64) slog[(size_t)b * (NH_*T_) + c * BS_ + tid] = lse_s[tid];
}

// ---------------------------------------------------------------------------
// Kernel 5: undo sort + softmax-combine the two hash rounds, scatter to (N,T,D)
// ---------------------------------------------------------------------------
__global__ __launch_bounds__(64) void combine_rounds(
    const float* __restrict__ so, const float* __restrict__ slog,
    const int* __restrict__ undo, float* __restrict__ out) {
  int t = blockIdx.x, b = blockIdx.y, f = threadIdx.x;
  const int* ub = undo + (size_t)b * (NH_*T_);
  int u0 = ub[t], u1 = ub[T_ + t];
  const float* lg = slog + (size_t)b * (NH_*T_);
  float l0 = lg[u0], l1 = lg[u1];
  float m = fmaxf(l0, l1);
  float e0 = expf(l0 - m), e1 = expf(l1 - m);
  float inv = 1.f / (e0 + e1);
  const float* sob = so + (size_t)b * (NH_*T_) * DH_;
  float val = (e0 * sob[(size_t)u0 * DH_ + f] +
               e1 * sob[(size_t)u1 * DH_ + f]) * inv;
  int n = b >> 4, h = b & 15;
  out[((size_t)n * T_ + t) * D_ + h * DH_ + f] = val;
}

// ---------------------------------------------------------------------------
// Kernel 6: Keras LayerNormalization (eps=1e-3), in-place on d_out.
// ---------------------------------------------------------------------------
__global__ __launch_bounds__(256) void layernorm_inplace(
    float* __restrict__ out, const float* __restrict__ gamma,
    const float* __restrict__ beta) {
  __shared__ float red[256];
  int row = blockIdx.x, tid = threadIdx.x;
  float* p = out + (size_t)row * D_;
  float loc[4]; float s = 0.f;
  #pragma unroll
  for (int i = 0; i < 4; ++i) { loc[i] = p[tid + (i<<8)]; s += loc[i]; }
  red[tid] = s; __syncthreads();
  for (int off = 128; off > 0; off >>= 1) {
    if (tid < off) red[tid] += red[tid + off];
    __syncthreads();
  }
  float mu = red[0] * (1.f / D_);
  __syncthreads();
  float vs = 0.f;
  #pragma unroll
  for (int i = 0; i < 4; ++i) { float dd = loc[i] - mu; vs += dd*dd; }
  red[tid] = vs; __syncthreads();
  for (int off = 128; off > 0; off >>= 1) {
    if (tid < off) red[tid] += red[tid + off];
    __syncthreads();
  }
  float var = red[0] * (1.f / D_);
  float inv = rsqrtf(var + 1e-3f);
  #pragma unroll
  for (int i = 0; i < 4; ++i) {
    int d = tid + (i<<8);
    p[d] = (loc[i] - mu) * inv * gamma[d] + beta[d];
  }
}

// ---------------------------------------------------------------------------
extern "C" void kernel_launch(void* const* d_in, const int* in_sizes, int n_in,
                              void* d_out, int out_size, void* d_ws, size_t ws_size,
                              hipStream_t stream) {
  (void)in_sizes; (void)n_in; (void)out_size; (void)ws_size;
  const float* x     = (const float*)d_in[0];
  const float* Wq    = (const float*)d_in[1];
  const float* bq    = (const float*)d_in[2];
  const float* Wv    = (const float*)d_in[3];
  const float* bv    = (const float*)d_in[4];
  const float* gamma = (const float*)d_in[5];
  const float* beta  = (const float*)d_in[6];
  const float* rot   = (const float*)d_in[7];
  float* out = (float*)d_out;

  // workspace layout (~264 MB)
  float* qk   = (float*)d_ws;                          // B*T*DH      = 64 MB
  float* vv   = qk + (size_t)B_ * T_ * DH_;            // 64 MB
  float* so   = vv + (size_t)B_ * T_ * DH_;            // B*NH*T*DH   = 128 MB
  float* slog = so + (size_t)B_ * NH_ * T_ * DH_;      // 2 MB
  int* buckets = (int*)(slog + (size_t)B_ * NH_ * T_); // 2 MB
  int* sticker = buckets + (size_t)B_ * NH_ * T_;      // 2 MB
  int* undo    = sticker + (size_t)B_ * NH_ * T_;      // 2 MB

  qv_gemm_f32_wmma<<<dim3(2048), dim3(256), 0, stream>>>(x, Wq, bq, Wv, bv, qk, vv);
  lsh_hash<<<dim3(T_/64, B_), dim3(64), 0, stream>>>(qk, rot, buckets);
  bucket_sort<<<dim3(B_), dim3(64), 0, stream>>>(buckets, sticker, undo);

  size_t shbytes = (size_t)ATTN_DWORDS * 4;   // ~118 KB < 320 KB/WGP
  (void)hipFuncSetAttribute((const void*)lsh_attn,
                            hipFuncAttributeMaxDynamicSharedMemorySize,
                            (int)shbytes);
  lsh_attn<<<dim3(C_, B_), dim3(128), shbytes, stream>>>(qk, vv, sticker, buckets, so, slog);
  combine_rounds<<<dim3(T_, B_), dim3(64), 0, stream>>>(so, slog, undo, out);
  layernorm_inplace<<<dim3(M_), dim3(256), 0, stream>>>(out, gamma, beta);
}